// CharRNN_86792699118080
// MI455X (gfx1250) — compile-verified
//
#include <hip/hip_runtime.h>
#include <hip/hip_bf16.h>

typedef __attribute__((ext_vector_type(2))) float v2f;
typedef __attribute__((ext_vector_type(8))) float v8f;

#define B_  256
#define L_  1024
#define V_  40
#define H_  128
#define VP  48            // vocab padded to 3 WMMA n-tiles
#define HP  (H_ + 4)      // h row padded by 4 dwords -> conflict-free A-fragment b64 loads
#define KSTEPS (H_/4)     // 32 k-steps of the 16x16x4 f32 WMMA

__launch_bounds__(256, 1)
__global__ void charrnn_fused(const int*   __restrict__ x,
                              const float* __restrict__ hidden0,
                              const float* __restrict__ embedding,
                              const float* __restrict__ Wh,
                              const float* __restrict__ Wo,
                              const float* __restrict__ b_h,
                              const float* __restrict__ b_y,
                              float* __restrict__ logits,
                              float* __restrict__ final_hidden)
{
    __shared__ float whT[H_ * H_];              // 64KB: whT[k*H+n] = Wh[n][k]
    __shared__ float wofrag[3 * KSTEPS * 32 * 2]; // 24KB: Wo B-fragments, pre-swizzled
    __shared__ float emb[V_ * H_];              // 20KB embedding table
    __shared__ float hbuf[16 * HP];             // current h tile (padded rows)
    __shared__ float bh_s[H_];
    __shared__ float by_s[VP];
    __shared__ int   tok[16];

    const int tid     = threadIdx.x;
    const int lane    = tid & 31;
    const int wave    = tid >> 5;
    const int bt      = blockIdx.x;         // 16-batch tile
    const int lane_lo = lane & 15;
    const bool hi     = lane >= 16;

    // ---------------- prologue: stage loop-invariants into LDS ----------------
    for (int i = tid; i < H_ * H_; i += 256) {
        int n = i >> 7, k = i & (H_ - 1);
        whT[k * H_ + n] = Wh[i];            // transpose Wh -> WhT[k][n]
    }
    for (int i = tid; i < V_ * H_; i += 256) emb[i]  = embedding[i];
    for (int i = tid; i < H_;      i += 256) bh_s[i] = b_h[i];
    for (int i = tid; i < VP;      i += 256) by_s[i] = (i < V_) ? b_y[i] : 0.0f;

    // Wo B-fragments in exact WMMA B layout, packed per (vtile, kstep, lane):
    //   element c in {0,1}: WoT[k][v] with k = 4*j + (lane>=16 ? 2 : 0) + c, v = 16*w + (lane&15)
    for (int i = tid; i < 3 * KSTEPS * 32; i += 256) {
        int l = i & 31, j = (i >> 5) & (KSTEPS - 1), w = i / (KSTEPS * 32);
        int v = w * 16 + (l & 15);
        int k = 4 * j + ((l >= 16) ? 2 : 0);
        float e0 = 0.0f, e1 = 0.0f;
        if (v < V_) { e0 = Wo[v * H_ + k]; e1 = Wo[v * H_ + k + 1]; }
        wofrag[i * 2 + 0] = e0;
        wofrag[i * 2 + 1] = e1;
    }
    for (int i = tid; i < 16 * H_; i += 256) {
        int m = i >> 7, k = i & (H_ - 1);
        hbuf[m * HP + k] = hidden0[(bt * 16 + m) * H_ + k];
    }
    __syncthreads();

    // ------- preload this wave's time-invariant Wh B-fragments into VGPRs -------
    const int   nb = wave * 16 + lane_lo;   // this lane's column of h / row of Wh
    float bw[2 * KSTEPS];                   // 64 VGPRs of B-matrix data
    #pragma unroll
    for (int j = 0; j < KSTEPS; ++j) {
        int k = 4 * j + (hi ? 2 : 0);
        bw[2 * j + 0] = whT[k       * H_ + nb];
        bw[2 * j + 1] = whT[(k + 1) * H_ + nb];
    }
    const float bhreg = bh_s[nb];
    const int   vcol  = wave * 16 + lane_lo;            // vocab column (waves 0..2)
    const float byreg = (wave < 3) ? by_s[vcol] : 0.0f;

    const long long obase = (long long)(bt * 16) * L_ * V_;

    // ------------------------------ time loop ------------------------------
    for (int t = 0; t < L_; ++t) {
        if (tid < 16) tok[tid] = x[(bt * 16 + tid) * L_ + t];
        __syncthreads();                    // tokens ready; hbuf == h_{t-1}

        // C init: embedding[x_t] + b_h   (C/D layout: VGPR r -> M = r (+8 for hi lanes))
        v8f c;
        #pragma unroll
        for (int r = 0; r < 8; ++r) {
            int m = r + (hi ? 8 : 0);
            c[r] = emb[tok[m] * H_ + nb] + bhreg;
        }

        // c += h_{t-1} @ Wh^T  on the matrix pipe (exact f32)
        #pragma unroll
        for (int j = 0; j < KSTEPS; ++j) {
            int k = 4 * j + (hi ? 2 : 0);
            const v2f a = *(const v2f*)&hbuf[lane_lo * HP + k]; // A[m=lane_lo][k,k+1]
            v2f b; b.x = bw[2 * j]; b.y = bw[2 * j + 1];
            c = __builtin_amdgcn_wmma_f32_16x16x4_f32(false, a, false, b,
                                                      (short)0, c, false, false);
        }

        // nonlinearity
        v8f h;
        #pragma unroll
        for (int r = 0; r < 8; ++r) h[r] = tanhf(c[r]);

        __syncthreads();                    // all A-reads of h_{t-1} complete
        #pragma unroll
        for (int r = 0; r < 8; ++r) {
            int m = r + (hi ? 8 : 0);
            hbuf[m * HP + nb] = h[r];
        }
        __syncthreads();                    // hbuf == h_t

        // fused vocab projection: logits[:, t, :] = h_t @ Wo^T + b_y (waves 0..2)
        if (wave < 3) {
            v8f d;
            #pragma unroll
            for (int r = 0; r < 8; ++r) d[r] = byreg;
            #pragma unroll
            for (int j = 0; j < KSTEPS; ++j) {
                int k = 4 * j + (hi ? 2 : 0);
                const v2f a = *(const v2f*)&hbuf[lane_lo * HP + k];
                const v2f b = *(const v2f*)&wofrag[((wave * KSTEPS + j) * 32 + lane) * 2];
                d = __builtin_amdgcn_wmma_f32_16x16x4_f32(false, a, false, b,
                                                          (short)0, d, false, false);
            }
            if (vcol < V_) {
                #pragma unroll
                for (int r = 0; r < 8; ++r) {
                    int m = r + (hi ? 8 : 0);
                    logits[obase + ((long long)m * L_ + t) * V_ + vcol] = d[r];
                }
            }
        }
    }

    __syncthreads();
    for (int i = tid; i < 16 * H_; i += 256) {
        int m = i >> 7, k = i & (H_ - 1);
        final_hidden[(bt * 16 + m) * H_ + k] = hbuf[m * HP + k];
    }
}

extern "C" void kernel_launch(void* const* d_in, const int* in_sizes, int n_in,
                              void* d_out, int out_size, void* d_ws, size_t ws_size,
                              hipStream_t stream) {
    const int*   x         = (const int*)  d_in[0];
    const float* hidden    = (const float*)d_in[1];
    const float* embedding = (const float*)d_in[2];
    const float* Wh        = (const float*)d_in[3];
    const float* Wo        = (const float*)d_in[4];
    const float* b_h       = (const float*)d_in[5];
    const float* b_y       = (const float*)d_in[6];

    float* logits       = (float*)d_out;                       // [B, L, V]
    float* final_hidden = logits + (size_t)B_ * L_ * V_;       // [B, H]

    charrnn_fused<<<B_ / 16, 256, 0, stream>>>(x, hidden, embedding, Wh, Wo,
                                               b_h, b_y, logits, final_hidden);
}